// LiftSplatShot_69286412419679
// MI455X (gfx1250) — compile-verified
//
#include <hip/hip_runtime.h>

// LiftSplatShot voxel pooling (splat) for MI455X / gfx1250.
//
// out[b, c, ix, iy] += x[p, c]  for every kept point p, where
//   (ix,iy,iz) = trunc((points[p] - pc_min) / dx), kept iff in [0,512)x[0,512)x{0}.
//
// Memory-bound scatter-add: ~222 MB reads + 128 MB zeroing + ~53M f32 atomics
// whose 128 MB target fits in the 192 MB global L2 (atomics resolve at L2).
// No matrix structure -> no WMMA. CDNA5-specific paths used:
//   - global_atomic_add_f32 scope:SCOPE_DEV (native f32 atomic, no CAS loop;
//     device scope since workgroups collide on voxels)
//   - global_store_b128 th:TH_STORE_WB for zeroing (keep zeros dirty in L2,
//     override wr-rinse -> saves a 128 MB HBM round trip before the atomics)
//   - global_prefetch_b8 ahead-prefetch of feature cachelines
//   - wave32 layout: one wave per point, 2 channels per lane.

#define NX0 512
#define NX1 512
#define NCH 64
#define PLANE (NX0 * NX1)          // 262144 floats per (b,c) plane

typedef __attribute__((ext_vector_type(4))) float v4f;

__global__ void lss_zero_out(float* __restrict__ out, int n4) {
    int t = blockIdx.x * blockDim.x + threadIdx.x;
    if (t < n4) {
        v4f z = {0.f, 0.f, 0.f, 0.f};
        float* p = out + (size_t)t * 4;
        // Keep the zeroed output resident + dirty in the 192MB L2 (WB hint
        // overrides the default wr-rinse policy) so the scatter's atomics hit.
        asm volatile("global_store_b128 %0, %1, off th:TH_STORE_WB"
                     :: "v"(p), "v"(z) : "memory");
    }
}

__global__ void lss_splat_scatter(const float* __restrict__ points,
                                  const float* __restrict__ xf,
                                  float* __restrict__ out,
                                  int P, int perBatch) {
    int t    = blockIdx.x * blockDim.x + threadIdx.x;
    int p    = t >> 5;          // one wave32 per point
    int lane = t & 31;          // lane handles channels {2*lane, 2*lane+1}
    if (p >= P) return;

    // p is wave-uniform: force it into an SGPR so the point-coordinate loads
    // and voxel math scalarize (SMEM + SALU instead of 32-wide VALU).
    int ps = __builtin_amdgcn_readfirstlane(p);

    // gfx1250 prefetch path: pull a future point's feature cachelines toward L2.
    __builtin_prefetch(xf + (size_t)(ps + 64) * NCH, 0, 1);

    const float* pp = points + (size_t)ps * 3;
    float px = pp[0], py = pp[1], pz = pp[2];

    // Match jnp.trunc((pt - pc_min)/dx).astype(int32): IEEE division (hipcc
    // default, no fast-math) followed by truncation toward zero ((int) cast).
    int ix = (int)((px + 51.2f) / 0.2f);
    int iy = (int)((py + 51.2f) / 0.2f);
    int iz = (int)((pz + 5.0f) / 8.0f);

    bool kept = (ix >= 0) & (ix < NX0) & (iy >= 0) & (iy < NX1) & (iz == 0);
    if (!kept) return;          // wave-uniform branch

    int b    = ps / perBatch;                      // batch index (B=2)
    int base = b * (NCH * PLANE) + ix * NX1 + iy;  // out[b, 0, ix, iy], flat

    // Coalesced: 32 lanes x float2 = the point's full 64-channel vector (256B).
    const float2 f = *(const float2*)(xf + (size_t)ps * NCH + lane * 2);

    float* o0 = out + base + (size_t)(lane * 2)     * PLANE;
    float* o1 = out + base + (size_t)(lane * 2 + 1) * PLANE;

    // Native no-return f32 atomic adds at device scope (cross-workgroup
    // accumulation). STOREcnt-tracked; s_endpgm performs implicit wait-idle.
    asm volatile("global_atomic_add_f32 %0, %1, off scope:SCOPE_DEV"
                 :: "v"(o0), "v"(f.x) : "memory");
    asm volatile("global_atomic_add_f32 %0, %1, off scope:SCOPE_DEV"
                 :: "v"(o1), "v"(f.y) : "memory");
}

extern "C" void kernel_launch(void* const* d_in, const int* in_sizes, int n_in,
                              void* d_out, int out_size, void* d_ws, size_t ws_size,
                              hipStream_t stream) {
    const float* points = (const float*)d_in[0];   // (B,N,D,H,W,3) f32
    const float* xf     = (const float*)d_in[1];   // (B,N,D,H,W,64) f32
    float*       out    = (float*)d_out;           // (2, 64, 512, 512) f32

    const int P        = in_sizes[1] / NCH;        // total points = 826,560
    const int perBatch = P / 2;                    // B = 2

    // 1) Clear output (harness poisons d_out); WB-hinted b128 stores.
    const int n4 = out_size / 4;
    lss_zero_out<<<(n4 + 255) / 256, 256, 0, stream>>>(out, n4);

    // 2) Scatter-add: one wave32 per point, 2 channels per lane.
    const long long threads = (long long)P * 32;
    const int blocks = (int)((threads + 255) / 256);
    lss_splat_scatter<<<blocks, 256, 0, stream>>>(points, xf, out, P, perBatch);
}